// VGMRF_banded_65085934404077
// MI455X (gfx1250) — compile-verified
//
#include <hip/hip_runtime.h>
#include <hip/hip_bf16.h>

#define BW    16
#define R_    16
#define KBAND 31   // 2*BW-1

typedef float v2f __attribute__((ext_vector_type(2)));
typedef float v8f __attribute__((ext_vector_type(8)));

// ---------------------------------------------------------------------------
// K1: banded Cholesky, sequential over N columns. One wave32.
// Ring buffer of last 15 produced band-columns in LDS (slot = col & 15).
// lane d (0..15) produces band element d of each column.
// ---------------------------------------------------------------------------
__global__ void __launch_bounds__(32)
chol_banded_kernel(const float* __restrict__ P, float* __restrict__ Lp, int n) {
  __shared__ float ring[16][17];
  const int lane = threadIdx.x;
  for (int i = lane; i < 16 * 17; i += 32) (&ring[0][0])[i] = 0.f;
  __syncthreads();
  const int d = lane & 15;
  for (int c = 0; c < n; ++c) {
    // s[d] = sum_{k=c-15..c-1} L[c-k][k] * L[c-k+d][k]
    float s = 0.f;
    #pragma unroll
    for (int a = 0; a < 15; ++a) {
      const int k = c - 1 - a;
      const int slot = k & 15;            // k<0 slots hold zeros (never written yet)
      const float diag = ring[slot][1 + a];        // broadcast read
      const int idx = 1 + a + d;
      const float e = (idx < 16) ? ring[slot][idx] : 0.f;
      s += diag * e;
    }
    const float r  = P[(size_t)d * n + c] - s;
    const float r0 = __shfl(r, 0, 32);
    const float iv = rsqrtf(r0);          // v = r / sqrt(r0); v[0] = sqrt(r0)
    const float v  = r * iv;
    if (lane < 16) {
      ring[c & 15][d] = v;                // overwrites col c-16 (no longer needed)
      Lp[(size_t)d * n + c] = v;
    }
    __syncthreads();
  }
}

// ---------------------------------------------------------------------------
// K2: backward solve L^T x = q_mu, 16 RHS handled by lanes 0..15. One wave32.
// Ring of last 15 solution rows in LDS.
// ---------------------------------------------------------------------------
__global__ void __launch_bounds__(32)
solve_lt_kernel(const float* __restrict__ Lp, const float* __restrict__ q_mu,
                float* __restrict__ m_out, int n) {
  __shared__ float xr[16][17];
  __shared__ float lcol[16];
  const int lane = threadIdx.x;
  for (int i = lane; i < 16 * 17; i += 32) (&xr[0][0])[i] = 0.f;
  __syncthreads();
  const int r = lane & 15;
  for (int j = n - 1; j >= 0; --j) {
    if (lane < 16) lcol[lane] = Lp[(size_t)lane * n + j];
    __syncthreads();
    float acc = q_mu[(size_t)j * R_ + r];
    #pragma unroll
    for (int t = 1; t <= 15; ++t)
      acc -= lcol[t] * xr[(j + t) & 15][r];   // slots beyond N-1 still hold zeros
    const float x = acc / lcol[0];
    if (lane < 16) {
      xr[j & 15][r] = x;
      m_out[(size_t)j * R_ + r] = x;
    }
    __syncthreads();
  }
}

// ---------------------------------------------------------------------------
// K3: band(Lp @ Qr) via FP32 WMMA. Both factors lower-banded width 16, so for
// output column block bj the band is covered by tiles (bj,bj), (bj+1,bj),
// (bj+2,bj):
//   C00 = Ad(bj)    * Bd
//   C10 = As(bj+1)  * Bd + Ad(bj+1) * Bs
//   C20 = As(bj+2)  * Bs
// => 4 tile-products = 16 x v_wmma_f32_16x16x4_f32 per wave.
// One wave per (column block, batch r). Tiles densified in LDS (pad 17).
// ---------------------------------------------------------------------------
__global__ void __launch_bounds__(32)
bandprod_wmma_kernel(const float* __restrict__ Lp, const float* __restrict__ Q,
                     float* __restrict__ Lqx, int n) {
  __shared__ float sA[4][16][17];   // Ad(bj), As(bj+1,bj), Ad(bj+1), As(bj+2,bj+1)
  __shared__ float sB[2][16][17];   // Bd(bj,bj), Bs(bj+1,bj)
  const int lane = threadIdx.x;
  const int bj   = blockIdx.x;
  const int rr   = blockIdx.y;
  const float* __restrict__ Qr = Q + (size_t)rr * BW * n;

  // Densify: tile(br,bc)[i][jl] = band[16*(br-bc)+i-jl][16*bc+jl] (0 outside)
  for (int idx = lane; idx < 6 * 256; idx += 32) {
    const int t  = idx >> 8;
    const int e  = idx & 255;
    const int i  = e >> 4;
    const int jl = e & 15;
    int delta, bc;
    const float* src;
    if (t < 4) { delta = t & 1;  bc = bj + (t >> 1); src = Lp; }
    else       { delta = t - 4;  bc = bj;            src = Qr; }
    const int col = 16 * bc + jl;
    const int dd  = 16 * delta + i - jl;
    float v = 0.f;
    if (col < n && dd >= 0 && dd < BW) v = src[(size_t)dd * n + col];
    if (t < 4) sA[t][i][jl] = v; else sB[t - 4][i][jl] = v;
  }
  __syncthreads();

  v8f c00 = {0.f, 0.f, 0.f, 0.f, 0.f, 0.f, 0.f, 0.f};
  v8f c10 = c00, c20 = c00;
  const int Mloc = lane & 15;       // A: row M / B: col N / C: N
  const int half = lane >> 4;
  #pragma unroll
  for (int ks = 0; ks < 4; ++ks) {
    const int k0 = 4 * ks + 2 * half;   // K pair per ISA 16x4 / 4x16 layouts
    v2f a0, a1, a2, a3, b0, b1;
    a0.x = sA[0][Mloc][k0]; a0.y = sA[0][Mloc][k0 + 1];
    a1.x = sA[1][Mloc][k0]; a1.y = sA[1][Mloc][k0 + 1];
    a2.x = sA[2][Mloc][k0]; a2.y = sA[2][Mloc][k0 + 1];
    a3.x = sA[3][Mloc][k0]; a3.y = sA[3][Mloc][k0 + 1];
    b0.x = sB[0][k0][Mloc]; b0.y = sB[0][k0 + 1][Mloc];
    b1.x = sB[1][k0][Mloc]; b1.y = sB[1][k0 + 1][Mloc];
    c00 = __builtin_amdgcn_wmma_f32_16x16x4_f32(false, a0, false, b0, (short)0, c00, false, false);
    c10 = __builtin_amdgcn_wmma_f32_16x16x4_f32(false, a1, false, b0, (short)0, c10, false, false);
    c10 = __builtin_amdgcn_wmma_f32_16x16x4_f32(false, a2, false, b1, (short)0, c10, false, false);
    c20 = __builtin_amdgcn_wmma_f32_16x16x4_f32(false, a3, false, b1, (short)0, c20, false, false);
  }

  // C layout: VGPR v -> M = v + 8*half, N = lane&15. Scatter band d = row-col.
  const size_t outBase = (size_t)rr * KBAND * n;
  const int col = 16 * bj + Mloc;
  #pragma unroll
  for (int v = 0; v < 8; ++v) {
    const int Mrow = v + 8 * half;
    const int d0 = Mrow - Mloc;       // tile (bj,  bj)
    const int d1 = 16 + d0;           // tile (bj+1,bj)
    const int d2 = 32 + d0;           // tile (bj+2,bj)
    if (d0 >= 0)  Lqx[outBase + (size_t)d0 * n + col] = c00[v];
    if (d1 <= 30) Lqx[outBase + (size_t)d1 * n + col] = c10[v];
    if (d2 <= 30) Lqx[outBase + (size_t)d2 * n + col] = c20[v];
  }
}

// ---------------------------------------------------------------------------
// K4: Takahashi recursion, one wave32 per batch r, reverse scan over N.
// W (30x30) lives in a ring-indexed 32x32 LDS array: the per-step shift
// Wn[1:,1:] = W[:29,:29] is pure index arithmetic (base decrements mod 32);
// only row 0 / col 0 are written each step.
// ---------------------------------------------------------------------------
__global__ void __launch_bounds__(32)
takahashi_kernel(const float* __restrict__ Lqx, float* __restrict__ Sx, int n) {
  __shared__ float W[32][33];
  __shared__ float lsh[KBAND];
  __shared__ float ush[30];
  const int lane = threadIdx.x;
  const int rr   = blockIdx.x;
  const float* __restrict__ Lr = Lqx + (size_t)rr * KBAND * n;
  for (int i = lane; i < 32 * 33; i += 32) (&W[0][0])[i] = 0.f;
  __syncthreads();
  int base = 0;
  for (int j = n - 1; j >= 0; --j) {
    if (lane < KBAND) {
      lsh[lane] = Lr[(size_t)lane * n + j];
      if (j > 0) __builtin_prefetch(&Lr[(size_t)lane * n + (j - 1)], 0, 3);
    }
    __syncthreads();
    const float inv_l0 = 1.f / lsh[0];
    // u = -(W @ l[1:]) / l0 ; lane i computes u[i]
    float u = 0.f;
    if (lane < 30) {
      float acc = 0.f;
      #pragma unroll 6
      for (int k = 0; k < 30; ++k)
        acc += W[(base + lane) & 31][(base + k) & 31] * lsh[1 + k];
      u = -acc * inv_l0;
      ush[lane] = u;
    }
    // sjj = (1/l0 - l[1:].u) / l0 via wave reduction
    float term = (lane < 30) ? lsh[1 + lane] * u : 0.f;
    #pragma unroll
    for (int off = 16; off > 0; off >>= 1)
      term += __shfl_down(term, off, 32);
    __syncthreads();                       // all W reads done; ush visible
    const int nb = (base + 31) & 31;       // base - 1 (mod 32)
    if (lane == 0) {
      const float sjj = (inv_l0 - term) * inv_l0;
      W[nb][nb] = sjj;
      Sx[(size_t)rr * n + j] = sjj;
    }
    if (lane < 29) {                       // Wn[0,1:30] and Wn[1:30,0] = u[:29]
      const float uv = ush[lane];
      const int p = (nb + 1 + lane) & 31;
      W[nb][p] = uv;
      W[p][nb] = uv;
    }
    base = nb;
    __syncthreads();
  }
}

// ---------------------------------------------------------------------------
extern "C" void kernel_launch(void* const* d_in, const int* in_sizes, int n_in,
                              void* d_out, int out_size, void* d_ws, size_t ws_size,
                              hipStream_t stream) {
  (void)n_in; (void)out_size; (void)ws_size;
  const float* P    = (const float*)d_in[0];   // (BW, N)
  const float* q_mu = (const float*)d_in[1];   // (N, R)
  const float* Q    = (const float*)d_in[2];   // (R, BW, N)
  float* out = (float*)d_out;

  const int n = in_sizes[0] / BW;              // N

  float* Lp  = (float*)d_ws;                   // BW * n floats
  float* Lqx = Lp + (size_t)BW * n;            // R_ * KBAND * n floats
  float* m_out  = out;                         // (n, R) — first output
  float* Sx_out = out + (size_t)n * R_;        // (R, n) — second output

  chol_banded_kernel  <<<1, 32, 0, stream>>>(P, Lp, n);
  bandprod_wmma_kernel<<<dim3(n / BW, R_), 32, 0, stream>>>(Lp, Q, Lqx, n);
  solve_lt_kernel     <<<1, 32, 0, stream>>>(Lp, q_mu, m_out, n);
  takahashi_kernel    <<<16, 32, 0, stream>>>(Lqx, Sx_out, n);
}